// ConstructAdjMatrixWithHomogeneous_24988119728841
// MI455X (gfx1250) — compile-verified
//
#include <hip/hip_runtime.h>
#include <math.h>

typedef float v2f __attribute__((ext_vector_type(2)));
typedef float v4f __attribute__((ext_vector_type(4)));
typedef float v8f __attribute__((ext_vector_type(8)));

#define EPSV 1e-9f

// ---------------- workspace init ----------------
__global__ __launch_bounds__(256)
void zero_ws_kernel(float* __restrict__ p, int n) {
  int i = blockIdx.x * blockDim.x + threadIdx.x;
  if (i < n) p[i] = 0.0f;
}

// ---------------- degree = adj @ ones via WMMA ----------------
// jobs: 0 rowsum(K)->deg[r], 1 rowsum(A)->deg[r],
//       2 colsum(A)->deg[nc+c], 3 rowsum(S)->deg[nc+r]
// One wave per block. A-operand tile is 16x4 f32; B = all-ones (layout
// invariant), so D[m,n] accumulates the 16 row sums. For column sums the
// ones matrix goes on the A side and the data tile on the B side.
__global__ __launch_bounds__(32)
void degree_wmma_kernel(const float* __restrict__ K, const float* __restrict__ A,
                        const float* __restrict__ S, const int* __restrict__ enable,
                        float* __restrict__ deg, float* __restrict__ total,
                        int nc, int nd, int nchunk)
{
  const int job = blockIdx.z;
  if ((job == 0 || job == 3) && (*enable == 0)) return;  // K/S blocks absent

  const float* M; int R, W, outBase;
  const bool colmode = (job == 2);
  if (job == 0)      { M = K; R = nc; W = nc; outBase = 0;  }
  else if (job == 1) { M = A; R = nc; W = nd; outBase = 0;  }
  else if (job == 2) { M = A; R = nc; W = nd; outBase = nc; }
  else               { M = S; R = nd; W = nd; outBase = nc; }

  const int redLen = colmode ? R : W;                 // reduction length
  const int quads  = (redLen + 15) >> 4;
  const int perChunk = ((quads + nchunk - 1) / nchunk) * 16;
  const int k0 = blockIdx.y * perChunk;
  int k1 = k0 + perChunk; if (k1 > redLen) k1 = redLen;
  if (k0 >= k1) return;

  const int lane = threadIdx.x;
  const int half = lane >> 4;   // which K-pair this lane supplies
  const int l15  = lane & 15;

  v8f acc = {};
  v2f ones; ones[0] = 1.0f; ones[1] = 1.0f;

  if (!colmode) {
    const int row = blockIdx.x * 16 + l15;
    const bool rv = (row < R);
    const bool tileFull = (blockIdx.x * 16 + 15 < R);
    const float* rp = M + (size_t)(rv ? row : 0) * W;
    if (tileFull && ((k1 - k0) & 15) == 0) {
      // fast path: no per-lane guards, 4 WMMAs / 4 b64 loads per iteration
      for (int k = k0; k < k1; k += 16) {
        v2f a0 = *(const v2f*)(rp + k      + 2 * half);
        v2f a1 = *(const v2f*)(rp + k + 4  + 2 * half);
        v2f a2 = *(const v2f*)(rp + k + 8  + 2 * half);
        v2f a3 = *(const v2f*)(rp + k + 12 + 2 * half);
        acc = __builtin_amdgcn_wmma_f32_16x16x4_f32(false, a0, false, ones, (short)0, acc, false, false);
        acc = __builtin_amdgcn_wmma_f32_16x16x4_f32(false, a1, false, ones, (short)0, acc, false, false);
        acc = __builtin_amdgcn_wmma_f32_16x16x4_f32(false, a2, false, ones, (short)0, acc, false, false);
        acc = __builtin_amdgcn_wmma_f32_16x16x4_f32(false, a3, false, ones, (short)0, acc, false, false);
      }
    } else {
      for (int k = k0; k < k1; k += 4) {
        v2f a = {};
        int kk = k + 2 * half;
        if (rv && kk < redLen)     a[0] = rp[kk];
        if (rv && kk + 1 < redLen) a[1] = rp[kk + 1];
        acc = __builtin_amdgcn_wmma_f32_16x16x4_f32(false, a, false, ones, (short)0, acc, false, false);
      }
    }
    // D layout: lane 0 holds rows 0..7 in acc[0..7], lane 16 rows 8..15
    if (l15 == 0) {
      float t = 0.0f;
      #pragma unroll
      for (int q = 0; q < 8; ++q) t += acc[q];
      #pragma unroll
      for (int q = 0; q < 8; ++q) {
        int r = blockIdx.x * 16 + half * 8 + q;
        if (r < R) atomicAdd(&deg[outBase + r], acc[q]);
      }
      atomicAdd(total, t);
    }
  } else {
    const int col = blockIdx.x * 16 + l15;
    const bool cv = (col < W);
    const bool tileFull = (blockIdx.x * 16 + 15 < W);
    const float* cp = M + (cv ? col : 0);
    if (tileFull && ((k1 - k0) & 7) == 0) {
      // fast path: half-wave-coalesced dword loads, 2 WMMAs per iteration
      for (int k = k0; k < k1; k += 8) {
        const size_t r0 = (size_t)(k + 2 * half) * W;
        const size_t r1 = (size_t)(k + 4 + 2 * half) * W;
        v2f b0; b0[0] = cp[r0]; b0[1] = cp[r0 + W];
        v2f b1; b1[0] = cp[r1]; b1[1] = cp[r1 + W];
        acc = __builtin_amdgcn_wmma_f32_16x16x4_f32(false, ones, false, b0, (short)0, acc, false, false);
        acc = __builtin_amdgcn_wmma_f32_16x16x4_f32(false, ones, false, b1, (short)0, acc, false, false);
      }
    } else {
      for (int k = k0; k < k1; k += 4) {
        v2f b = {};
        int kk = k + 2 * half;
        if (cv && kk < redLen)     b[0] = cp[(size_t)kk * W];
        if (cv && kk + 1 < redLen) b[1] = cp[(size_t)(kk + 1) * W];
        acc = __builtin_amdgcn_wmma_f32_16x16x4_f32(false, ones, false, b, (short)0, acc, false, false);
      }
    }
    // every row of D holds the column sums; lane n (n<16) acc[0] = colsum(col)
    if (half == 0 && cv) {
      atomicAdd(&deg[outBase + col], acc[0]);
      atomicAdd(total, acc[0]);
    }
  }
}

// ---------------- d = (deg+eps)^-1/2 ----------------
__global__ __launch_bounds__(256)
void finalize_deg_kernel(const float* __restrict__ deg, float* __restrict__ dval, int n) {
  int i = blockIdx.x * blockDim.x + threadIdx.x;
  if (i < n) dval[i] = 1.0f / sqrtf(deg[i] + EPSV);
}

// ---------------- diagonal blocks (K / S): streaming v4f ----------------
__global__ __launch_bounds__(256)
void diag_block_kernel(const float* __restrict__ M, const float* __restrict__ dval,
                       const float* __restrict__ total, const int* __restrict__ enable,
                       float* __restrict__ out, int W, int N, int off)
{
  int j4 = blockIdx.x * blockDim.x + threadIdx.x;
  int i = blockIdx.y;
  if (j4 >= (W >> 2)) return;
  int j = j4 << 2;
  v4f o;
  bool ident = (*total == 0.0f) || (*enable == 0);
  if (ident) {
    o.x = (i == j    ) ? 1.f : 0.f;
    o.y = (i == j + 1) ? 1.f : 0.f;
    o.z = (i == j + 2) ? 1.f : 0.f;
    o.w = (i == j + 3) ? 1.f : 0.f;
  } else {
    float di = dval[off + i];
    const v4f m = *(const v4f*)(M + (size_t)i * W + j);
    o.x = di * m.x * dval[off + j    ] + ((i == j    ) ? 1.f : 0.f);
    o.y = di * m.y * dval[off + j + 1] + ((i == j + 1) ? 1.f : 0.f);
    o.z = di * m.z * dval[off + j + 2] + ((i == j + 2) ? 1.f : 0.f);
    o.w = di * m.w * dval[off + j + 3] + ((i == j + 3) ? 1.f : 0.f);
  }
  // output is write-once / never re-read: non-temporal store keeps the
  // 192MB of inputs resident in the 192MB L2 between passes
  __builtin_nontemporal_store(o, (v4f*)(out + (size_t)(off + i) * N + off + j));
}

// ---------------- A / A^T blocks: read A once, LDS transpose ----------------
__global__ __launch_bounds__(256)
void cross_block_kernel(const float* __restrict__ A, const float* __restrict__ dval,
                        const float* __restrict__ total, float* __restrict__ out,
                        int nc, int nd, int N)
{
  __shared__ float tile[32][33];           // +1 pad: conflict-free transpose
  const int tx = threadIdx.x;              // 0..7  -> 8 x v4f = 32 cols
  const int ty = threadIdx.y;              // 0..31 -> 32 rows
  const int c0 = blockIdx.x * 32;
  const int i0 = blockIdx.y * 32;
  const bool zero = (*total == 0.0f);

  const int i = i0 + ty;
  const int c = c0 + (tx << 2);

  v4f v = {};
  const bool inA = (i < nc) && (c + 3 < nd);
  if (!zero && inA) {
    const v4f a = *(const v4f*)(A + (size_t)i * nd + c);
    const float di = dval[i];
    v.x = di * a.x * dval[nc + c    ];
    v.y = di * a.y * dval[nc + c + 1];
    v.z = di * a.z * dval[nc + c + 2];
    v.w = di * a.w * dval[nc + c + 3];
  }
  if (inA)  // top-right block, coalesced
    __builtin_nontemporal_store(v, (v4f*)(out + (size_t)i * N + nc + c));

  tile[ty][(tx << 2) + 0] = v.x;
  tile[ty][(tx << 2) + 1] = v.y;
  tile[ty][(tx << 2) + 2] = v.z;
  tile[ty][(tx << 2) + 3] = v.w;
  __syncthreads();

  // bottom-left block: out[nc+c', i'] = out[i', nc+c'] (symmetric scaling)
  const int rb = nc + c0 + ty;
  const int cb = i0 + (tx << 2);
  v4f w;
  w.x = tile[(tx << 2) + 0][ty];
  w.y = tile[(tx << 2) + 1][ty];
  w.z = tile[(tx << 2) + 2][ty];
  w.w = tile[(tx << 2) + 3][ty];
  if ((c0 + ty) < nd && (cb + 3) < nc)
    __builtin_nontemporal_store(w, (v4f*)(out + (size_t)rb * N + cb));
}

extern "C" void kernel_launch(void* const* d_in, const int* in_sizes, int n_in,
                              void* d_out, int out_size, void* d_ws, size_t ws_size,
                              hipStream_t stream)
{
  const float* K = (const float*)d_in[0];      // filtered_cell_kernel  [nc,nc]
  const float* S = (const float*)d_in[1];      // filtered_drug_sim     [nd,nd]
  const float* A = (const float*)d_in[2];      // original_cell_drug_adj[nc,nd]
  const int* enable = (const int*)d_in[3];     // scalar flag
  float* out = (float*)d_out;

  const int nc = (int)(sqrt((double)in_sizes[0]) + 0.5);
  const int nd = (int)(sqrt((double)in_sizes[1]) + 0.5);
  const int N = nc + nd;

  float* wsF   = (float*)d_ws;
  float* deg   = wsF;          // N floats
  float* total = wsF + N;      // 1 float
  float* dval  = wsF + N + 1;  // N floats

  // 1) zero deg + total (workspace is poisoned, must re-init every call)
  zero_ws_kernel<<<dim3((N + 1 + 255) / 256), dim3(256), 0, stream>>>(wsF, N + 1);

  // 2) degrees via WMMA GEMV against all-ones
  const int NCHUNK = 10;
  const int maxn = nc > nd ? nc : nd;
  dim3 gdeg((maxn + 15) / 16, NCHUNK, 4);
  degree_wmma_kernel<<<gdeg, dim3(32), 0, stream>>>(K, A, S, enable, deg, total,
                                                    nc, nd, NCHUNK);

  // 3) d = (deg + eps)^-1/2
  finalize_deg_kernel<<<dim3((N + 255) / 256), dim3(256), 0, stream>>>(deg, dval, N);

  // 4) top-left (K) and bottom-right (S) blocks
  {
    dim3 g(((nc >> 2) + 255) / 256, nc);
    diag_block_kernel<<<g, dim3(256), 0, stream>>>(K, dval, total, enable, out, nc, N, 0);
  }
  {
    dim3 g(((nd >> 2) + 255) / 256, nd);
    diag_block_kernel<<<g, dim3(256), 0, stream>>>(S, dval, total, enable, out, nd, N, nc);
  }

  // 5) A and A^T blocks (single read of A, LDS transpose for the mirror)
  {
    dim3 b(8, 32);
    dim3 g((nd + 31) / 32, (nc + 31) / 32);
    cross_block_kernel<<<g, b, 0, stream>>>(A, dval, total, out, nc, nd, N);
  }
}